// S4DKernel_4853313044573
// MI455X (gfx1250) — compile-verified
//
#include <hip/hip_runtime.h>

// Problem dims (match reference exactly).
constexpr int Bdim = 4;
constexpr int Cdim = 1024;   // channels
constexpr int Ndim = 64;     // modes
constexpr int Tdim = 2048;   // sequence length

typedef float v2f __attribute__((ext_vector_type(2)));
typedef float v4f __attribute__((ext_vector_type(4)));
typedef float v8f __attribute__((ext_vector_type(8)));

__device__ __forceinline__ float softplus_fast(float x) {
    // max(x,0) + log(1 + exp(-|x|)) using fast v_exp_f32 / v_log_f32.
    return fmaxf(x, 0.0f) + __logf(1.0f + __expf(-fabsf(x)));
}

// ---------------------------------------------------------------------------
// Kernel 1: K = mix @ E  via V_WMMA_F32_16X16X4_F32.
//   E[j,t] = exp(-rate[j] * t),  rate[j] = softplus(log_tau[j])*softplus(log_dt).
// rate[] is computed once per block into LDS (64 scalars); the WMMA loop then
// needs only v_mul + v_exp per B element. One wave computes one 16(ch)x16(t)
// tile with 16 chained WMMAs (K-dim = 64). Block = 256 threads = 8 waves
// stacked along channels. Grid = (T/16, C/16/8) = (128, 8).
// ---------------------------------------------------------------------------
__global__ void __launch_bounds__(256)
kgen_wmma(const float* __restrict__ mix,
          const float* __restrict__ log_tau,
          const float* __restrict__ log_dt,
          float* __restrict__ Kout) {
    __shared__ float sRate[Ndim];

    if (threadIdx.x < Ndim) {
        const float dtv = softplus_fast(log_dt[0]);
        sRate[threadIdx.x] = softplus_fast(log_tau[threadIdx.x]) * dtv;
    }
    __syncthreads();   // EXEC fully re-converged before any WMMA

    const int lane = threadIdx.x & 31;
    const int wave = threadIdx.x >> 5;
    const int row  = lane & 15;   // M-row (A) / N-col (B,C) index for this lane
    const int half = lane >> 4;   // which half-wave

    const int t0 = blockIdx.x * 16;
    const int c0 = (blockIdx.y * 8 + wave) * 16;

    const float tB = (float)(t0 + row);      // B/C column -> time index

    v8f acc = {};
    const float* mixRow = mix + (size_t)(c0 + row) * Ndim;

    #pragma unroll
    for (int j0 = 0; j0 < Ndim; j0 += 4) {
        const int ja = j0 + half * 2;        // this half-wave's two K-cols / B-rows

        // A-tile (16x4 f32): lane holds mix[row, ja], mix[row, ja+1]
        v2f a;
        a.x = mixRow[ja];
        a.y = mixRow[ja + 1];

        // B-tile (4x16 f32): lane holds E[ja, t], E[ja+1, t]
        v2f b;
        b.x = __expf(-sRate[ja]     * tB);
        b.y = __expf(-sRate[ja + 1] * tB);

        acc = __builtin_amdgcn_wmma_f32_16x16x4_f32(
            /*neg_a=*/false, a, /*neg_b=*/false, b,
            /*c_mod=*/(short)0, acc, /*reuse_a=*/false, /*reuse_b=*/false);
    }

    // C/D layout: VGPR r -> M = r (lanes 0-15) or M = 8+r (lanes 16-31); N = lane&15
    #pragma unroll
    for (int r = 0; r < 8; ++r) {
        const int m = half * 8 + r;
        Kout[(size_t)(c0 + m) * Tdim + (t0 + row)] = acc[r];
    }
}

// ---------------------------------------------------------------------------
// Kernel 2: causal depthwise conv + skip:
//   y[b,c,t] = sum_{j=0..t} K[c,j] * u[b,c,t-j] + D[c]*u[b,c,t]
// One block per (b,c); u[b,c,:] and K[c,:] staged in LDS via b128 vectors.
// Each thread produces 8 consecutive t's (tb = tid*8). Main loop processes
// j in groups of 4 with a rolling 12-wide u window held in three float4
// registers: per group just TWO ds_load_b128 (new u quad + 4 kernel taps)
// feed 32 FMAs. Tail handles j in [tb, tb+7] with causal guards.
// ---------------------------------------------------------------------------
__global__ void __launch_bounds__(256)
kconv(const float* __restrict__ u,
      const float* __restrict__ Dv,
      const float* __restrict__ Kc,
      float* __restrict__ y) {
    __shared__ float sU[Tdim];
    __shared__ float sK[Tdim];

    const int c = blockIdx.x;
    const int b = blockIdx.y;

    const float* uc = u  + ((size_t)b * Cdim + c) * Tdim;
    const float* kc = Kc + (size_t)c * Tdim;

    // Vectorized staging: 2048 floats = 512 float4 per array, 2 per thread.
    for (int i = threadIdx.x; i < Tdim / 4; i += 256) {
        ((v4f*)sU)[i] = ((const v4f*)uc)[i];
        ((v4f*)sK)[i] = ((const v4f*)kc)[i];
    }
    __syncthreads();

    const float d  = Dv[c];
    const int   tb = threadIdx.x << 3;    // first of 8 consecutive outputs

    float acc[8];
    #pragma unroll
    for (int s = 0; s < 8; ++s) acc[s] = d * sU[tb + s];

    // Rolling window over u: for group jg (j = jg..jg+3, all < tb), we need
    // u[base-3 .. base+7] with base = tb - jg.  Window regs:
    //   w0 = u[base-4 .. base-1], w1 = u[base .. base+3], w2 = u[base+4 .. base+7]
    // Indexed as i = (s - q) + 4 in [1,11] for output s, tap q.
    if (tb > 0) {
        v4f w1 = *(const v4f*)&sU[tb];       // base = tb for first group
        v4f w2 = *(const v4f*)&sU[tb + 4];
        for (int jg = 0; jg < tb; jg += 4) {
            const int base = tb - jg;
            const v4f w0 = *(const v4f*)&sU[base - 4];
            const v4f k4 = *(const v4f*)&sK[jg];
            #pragma unroll
            for (int q = 0; q < 4; ++q) {
                #pragma unroll
                for (int s = 0; s < 8; ++s) {
                    const int i = s - q + 4;           // 1..11, compile-time
                    const float uv = (i < 4) ? w0[i] : (i < 8) ? w1[i - 4] : w2[i - 8];
                    acc[s] += k4[q] * uv;
                }
            }
            w2 = w1;   // slide window down by 4 for next group
            w1 = w0;
        }
    }

    // Tail: j = tb+e, e in [0,7]; contributes only to outputs s >= e,
    // with u index t - j = s - e in [0,7].
    {
        const v4f u0 = *(const v4f*)&sU[0];
        const v4f u1 = *(const v4f*)&sU[4];
        #pragma unroll
        for (int e = 0; e < 8; ++e) {
            const float kj = sK[tb + e];
            #pragma unroll
            for (int s = e; s < 8; ++s) {
                const int i = s - e;                   // 0..7, compile-time
                const float uv = (i < 4) ? u0[i] : u1[i - 4];
                acc[s] += kj * uv;
            }
        }
    }

    float* yc = y + ((size_t)b * Cdim + c) * Tdim;
    v4f o0, o1;
    #pragma unroll
    for (int s = 0; s < 4; ++s) { o0[s] = acc[s]; o1[s] = acc[s + 4]; }
    *(v4f*)&yc[tb]     = o0;
    *(v4f*)&yc[tb + 4] = o1;
}

// ---------------------------------------------------------------------------
// Host-side launcher. Inputs (setup_inputs order):
//   d_in[0]=u (B,C,T) f32, d_in[1]=log_tau (N) f32, d_in[2]=mix (C,N) f32,
//   d_in[3]=D (C) f32, d_in[4]=log_dt (1) f32.
// d_out: (B,C,T) f32.  d_ws: C*T*4 = 8 MB for K (fully overwritten each call).
// ---------------------------------------------------------------------------
extern "C" void kernel_launch(void* const* d_in, const int* in_sizes, int n_in,
                              void* d_out, int out_size, void* d_ws, size_t ws_size,
                              hipStream_t stream) {
    const float* u       = (const float*)d_in[0];
    const float* log_tau = (const float*)d_in[1];
    const float* mix     = (const float*)d_in[2];
    const float* Dv      = (const float*)d_in[3];
    const float* log_dt  = (const float*)d_in[4];
    float*       y       = (float*)d_out;
    float*       Kws     = (float*)d_ws;

    (void)in_sizes; (void)n_in; (void)out_size; (void)ws_size;

    dim3 g1(Tdim / 16, (Cdim / 16) / 8);     // (128, 8)
    kgen_wmma<<<g1, 256, 0, stream>>>(mix, log_tau, log_dt, Kws);

    dim3 g2(Cdim, Bdim);                     // (1024, 4)
    kconv<<<g2, 256, 0, stream>>>(u, Dv, Kws, y);
}